// _PointnetSAModuleFSBase_11974368821850
// MI455X (gfx1250) — compile-verified
//
#include <hip/hip_runtime.h>

#define BB 2
#define NN 16384
#define CIN 64
#define NPT 4096
#define NS 32
#define R2 0.64f

typedef __attribute__((ext_vector_type(2))) float v2f;
typedef __attribute__((ext_vector_type(8))) float v8f;

__device__ inline v8f wmma_f32(v2f a, v2f b, v8f c) {
  // V_WMMA_F32_16X16X4_F32 : D(16x16 f32) = A(16x4) * B(4x16) + C
  return __builtin_amdgcn_wmma_f32_16x16x4_f32(
      false, a, false, b, (short)0, c, false, false);
}

// ---------------------------------------------------------------------------
// Kernel 0: transpose features (B, 64, N) -> featT (B, N, 64)
// ---------------------------------------------------------------------------
__global__ __launch_bounds__(256) void transpose_feat(const float* __restrict__ f,
                                                      float* __restrict__ ft) {
  __shared__ float tile[32][33];
  int b = blockIdx.z;
  int c0 = blockIdx.y * 32;
  int n0 = blockIdx.x * 32;
  int tx = threadIdx.x, ty = threadIdx.y;
  for (int r = ty; r < 32; r += 8)
    tile[r][tx] = f[((size_t)b * CIN + (c0 + r)) * NN + n0 + tx];
  __syncthreads();
  for (int r = ty; r < 32; r += 8)
    ft[((size_t)b * NN + n0 + r) * CIN + c0 + tx] = tile[tx][r];
}

// ---------------------------------------------------------------------------
// Kernel 1: furthest point sampling. One 1024-thread block per batch.
// Each thread owns 16 points (p = tid + 1024*k): coords + mind live in VGPRs.
// Writes new_xyz (first section of d_out).
// ---------------------------------------------------------------------------
__global__ __launch_bounds__(1024) void fps_kernel(const float* __restrict__ xyz,
                                                   float* __restrict__ new_xyz) {
  int b = blockIdx.x;
  int t = threadIdx.x;
  const float* xb = xyz + (size_t)b * NN * 3;

  __shared__ int s_idx[NPT];
  __shared__ float rv[32];
  __shared__ int ri[32];
  __shared__ int s_last;

  float px[16], py[16], pz[16], mind[16];
#pragma unroll
  for (int k = 0; k < 16; k++) {
    int p = t + 1024 * k;
    px[k] = xb[p * 3 + 0];
    py[k] = xb[p * 3 + 1];
    pz[k] = xb[p * 3 + 2];
    mind[k] = 1e10f;
  }

  int last = 0;
  for (int i = 0; i < NPT; i++) {
    float lx = xb[last * 3 + 0];
    float ly = xb[last * 3 + 1];
    float lz = xb[last * 3 + 2];
    float best = -1.0f;
    int bidx = 0;
#pragma unroll
    for (int k = 0; k < 16; k++) {
      float dx = px[k] - lx, dy = py[k] - ly, dz = pz[k] - lz;
      float d = dx * dx + dy * dy + dz * dz;
      float m = fminf(mind[k], d);
      mind[k] = m;
      if (m > best) { best = m; bidx = t + 1024 * k; }
    }
    // wave32 reduce (max value, min index on ties)
#pragma unroll
    for (int off = 16; off > 0; off >>= 1) {
      float ov = __shfl_down(best, off, 32);
      int oi = __shfl_down(bidx, off, 32);
      if (ov > best || (ov == best && oi < bidx)) { best = ov; bidx = oi; }
    }
    if ((t & 31) == 0) { rv[t >> 5] = best; ri[t >> 5] = bidx; }
    __syncthreads();
    if (t < 32) {
      best = rv[t];
      bidx = ri[t];
#pragma unroll
      for (int off = 16; off > 0; off >>= 1) {
        float ov = __shfl_down(best, off, 32);
        int oi = __shfl_down(bidx, off, 32);
        if (ov > best || (ov == best && oi < bidx)) { best = ov; bidx = oi; }
      }
      if (t == 0) { s_idx[i] = last; s_last = bidx; }
    }
    __syncthreads();
    last = s_last;
  }

  __syncthreads();
  for (int j = t; j < NPT; j += 1024) {
    int p = s_idx[j];
    new_xyz[((size_t)b * NPT + j) * 3 + 0] = xb[p * 3 + 0];
    new_xyz[((size_t)b * NPT + j) * 3 + 1] = xb[p * 3 + 1];
    new_xyz[((size_t)b * NPT + j) * 3 + 2] = xb[p * 3 + 2];
  }
}

// ---------------------------------------------------------------------------
// Kernel 2: ball query. One wave per center; first NS in-range points in order.
// ---------------------------------------------------------------------------
__global__ __launch_bounds__(256) void ballquery_kernel(const float* __restrict__ xyz,
                                                        const float* __restrict__ new_xyz,
                                                        int* __restrict__ gidx,
                                                        int* __restrict__ gcnt) {
  int gtid = blockIdx.x * blockDim.x + threadIdx.x;
  int g = gtid >> 5;            // center id
  int lane = gtid & 31;
  int b = g / NPT;
  int m = g - b * NPT;

  const float* xb = xyz + (size_t)b * NN * 3;
  const float* c = new_xyz + ((size_t)b * NPT + m) * 3;
  float cx = c[0], cy = c[1], cz = c[2];

  int* base = gidx + (size_t)g * NS;
  int cnt = 0;
  int firstp = 0;

  for (int ch = 0; ch < NN / 32; ch++) {
    int p = ch * 32 + lane;
    float dx = xb[p * 3 + 0] - cx;
    float dy = xb[p * 3 + 1] - cy;
    float dz = xb[p * 3 + 2] - cz;
    float d2 = dx * dx + dy * dy + dz * dz;
    bool win = d2 <= R2;
    unsigned mask = (unsigned)__ballot(win);
    if (cnt == 0 && mask) firstp = ch * 32 + (__ffs(mask) - 1);
    int rank = cnt + __popc(mask & ((1u << lane) - 1u));
    if (win && rank < NS) base[rank] = p;
    cnt += __popc(mask);
    if (cnt >= NS) break;
  }
  cnt = min(cnt, NS);
  if (cnt < NS && lane >= cnt) base[lane] = (cnt > 0) ? firstp : 0;
  if (lane == 0) gcnt[g] = cnt;
}

// ---------------------------------------------------------------------------
// Kernel 3: gather + 3-layer MLP (WMMA f32 16x16x4) + ReLU + masked max-pool.
// One wave per center (32 samples = two 16-row A tiles). 2 waves / block.
// Staged channel order: [feat 0..63 | xyz 64..66 | pad 67].
// ---------------------------------------------------------------------------
__device__ inline float w1_at(const float* __restrict__ w1, int col, int k) {
  if (k < 64) return w1[col * 67 + 3 + k];   // feature channels
  if (k < 67) return w1[col * 67 + (k - 64)]; // xyz channels
  return 0.0f;                                // pad
}

__global__ __launch_bounds__(64) void mlp_pool_kernel(
    const float* __restrict__ xyz, const float* __restrict__ new_xyz,
    const float* __restrict__ featT, const int* __restrict__ gidx,
    const int* __restrict__ gcnt, const float* __restrict__ w1,
    const float* __restrict__ b1, const float* __restrict__ w2,
    const float* __restrict__ b2, const float* __restrict__ w3,
    const float* __restrict__ b3, float* __restrict__ outf) {
  __shared__ __align__(16) float s0[2][32 * 68];
  __shared__ __align__(16) float s1[2][32 * 64];

  int wv = threadIdx.x >> 5;
  int lane = threadIdx.x & 31;
  int lcol = lane & 15;
  int khalf = (lane >> 4) * 2;   // lanes 16-31 take K+2,K+3
  float* st0 = s0[wv];
  float* st1 = s1[wv];

  int g = blockIdx.x * 2 + wv;
  int b = g / NPT;
  int m = g - b * NPT;

  // ---- gather 32 points into st0 ----
  int p = gidx[(size_t)g * NS + lane];
  int cnt = gcnt[g];
  const float* xb = xyz + (size_t)b * NN * 3;
  const float* c = new_xyz + ((size_t)b * NPT + m) * 3;
  st0[lane * 68 + 64] = xb[p * 3 + 0] - c[0];
  st0[lane * 68 + 65] = xb[p * 3 + 1] - c[1];
  st0[lane * 68 + 66] = xb[p * 3 + 2] - c[2];
  st0[lane * 68 + 67] = 0.0f;
  const float4* fr = reinterpret_cast<const float4*>(featT + ((size_t)b * NN + p) * CIN);
  float4* dst = reinterpret_cast<float4*>(st0 + lane * 68);
#pragma unroll
  for (int v = 0; v < 16; v++) dst[v] = fr[v];

  // ---- layer 1: 68 -> 64 ----
  for (int ct = 0; ct < 4; ct++) {
    int col = ct * 16 + lcol;
    float bv = b1[col];
    v8f acc0 = {bv, bv, bv, bv, bv, bv, bv, bv};
    v8f acc1 = acc0;
#pragma unroll
    for (int k0 = 0; k0 < 68; k0 += 4) {
      int kk = k0 + khalf;
      v2f a0 = {st0[lcol * 68 + kk], st0[lcol * 68 + kk + 1]};
      v2f a1 = {st0[(16 + lcol) * 68 + kk], st0[(16 + lcol) * 68 + kk + 1]};
      v2f bf = {w1_at(w1, col, kk), w1_at(w1, col, kk + 1)};
      acc0 = wmma_f32(a0, bf, acc0);
      acc1 = wmma_f32(a1, bf, acc1);
    }
#pragma unroll
    for (int j = 0; j < 8; j++) {
      int r = (lane < 16) ? j : j + 8;
      st1[r * 64 + col] = fmaxf(acc0[j], 0.0f);
      st1[(16 + r) * 64 + col] = fmaxf(acc1[j], 0.0f);
    }
  }

  // ---- layer 2: 64 -> 64 ----
  for (int ct = 0; ct < 4; ct++) {
    int col = ct * 16 + lcol;
    float bv = b2[col];
    v8f acc0 = {bv, bv, bv, bv, bv, bv, bv, bv};
    v8f acc1 = acc0;
#pragma unroll
    for (int k0 = 0; k0 < 64; k0 += 4) {
      int kk = k0 + khalf;
      v2f a0 = {st1[lcol * 64 + kk], st1[lcol * 64 + kk + 1]};
      v2f a1 = {st1[(16 + lcol) * 64 + kk], st1[(16 + lcol) * 64 + kk + 1]};
      v2f bf = {w2[col * 64 + kk], w2[col * 64 + kk + 1]};
      acc0 = wmma_f32(a0, bf, acc0);
      acc1 = wmma_f32(a1, bf, acc1);
    }
#pragma unroll
    for (int j = 0; j < 8; j++) {
      int r = (lane < 16) ? j : j + 8;
      st0[r * 68 + col] = fmaxf(acc0[j], 0.0f);
      st0[(16 + r) * 68 + col] = fmaxf(acc1[j], 0.0f);
    }
  }

  // ---- layer 3: 64 -> 128, fused ReLU + max-pool over 32 samples ----
  for (int ct = 0; ct < 8; ct++) {
    int col = ct * 16 + lcol;
    float bv = b3[col];
    v8f acc0 = {bv, bv, bv, bv, bv, bv, bv, bv};
    v8f acc1 = acc0;
#pragma unroll
    for (int k0 = 0; k0 < 64; k0 += 4) {
      int kk = k0 + khalf;
      v2f a0 = {st0[lcol * 68 + kk], st0[lcol * 68 + kk + 1]};
      v2f a1 = {st0[(16 + lcol) * 68 + kk], st0[(16 + lcol) * 68 + kk + 1]};
      v2f bf = {w3[col * 64 + kk], w3[col * 64 + kk + 1]};
      acc0 = wmma_f32(a0, bf, acc0);
      acc1 = wmma_f32(a1, bf, acc1);
    }
    float pool = 0.0f;  // ReLU floor == 0, so max with 0 implements relu+pool
#pragma unroll
    for (int j = 0; j < 8; j++) {
      pool = fmaxf(pool, acc0[j]);
      pool = fmaxf(pool, acc1[j]);
    }
    pool = fmaxf(pool, __shfl_xor(pool, 16, 32));
    if (cnt == 0) pool = 0.0f;
    if (lane < 16) outf[((size_t)b * 128 + col) * NPT + m] = pool;
  }
}

// ---------------------------------------------------------------------------
extern "C" void kernel_launch(void* const* d_in, const int* in_sizes, int n_in,
                              void* d_out, int out_size, void* d_ws, size_t ws_size,
                              hipStream_t stream) {
  const float* xyz  = (const float*)d_in[0];
  const float* feat = (const float*)d_in[1];
  const float* w1   = (const float*)d_in[2];
  const float* b1   = (const float*)d_in[3];
  const float* w2   = (const float*)d_in[4];
  const float* b2   = (const float*)d_in[5];
  const float* w3   = (const float*)d_in[6];
  const float* b3   = (const float*)d_in[7];

  float* out = (float*)d_out;
  float* new_xyz = out;                          // (B, NPT, 3)
  float* out_feat = out + (size_t)BB * NPT * 3;  // (B, 128, NPT)

  float* featT = (float*)d_ws;                                  // B*N*64 f32
  int* gidx = (int*)((char*)d_ws + (size_t)BB * NN * CIN * 4);  // B*NPT*NS i32
  int* gcnt = (int*)((char*)gidx + (size_t)BB * NPT * NS * 4);  // B*NPT i32

  dim3 tb(32, 8);
  dim3 tg(NN / 32, CIN / 32, BB);
  transpose_feat<<<tg, tb, 0, stream>>>(feat, featT);

  fps_kernel<<<BB, 1024, 0, stream>>>(xyz, new_xyz);

  ballquery_kernel<<<(BB * NPT) / 8, 256, 0, stream>>>(xyz, new_xyz, gidx, gcnt);

  mlp_pool_kernel<<<(BB * NPT) / 2, 64, 0, stream>>>(
      xyz, new_xyz, featT, gidx, gcnt, w1, b1, w2, b2, w3, b3, out_feat);
}